// SimpleScheduler_H_34213709480231
// MI455X (gfx1250) — compile-verified
//
#include <hip/hip_runtime.h>
#include <hip/hip_bf16.h>
#include <math.h>

// ---------------------------------------------------------------------------
// Problem constants (from reference)
// ---------------------------------------------------------------------------
#define HIDDEN     4096
#define NOUT       1536          // NUM_SUB * H * 2
#define BATCH      16384
#define NUM_PREFIX 8
#define NUM_SUB    24
#define NHEADS     32
#define NGROUPS    384           // NUM_SUB * H / RANK
#define TAU_F      5.0f
#define OUT_PER_B  (32 * 2 * 32) // (layers, 2, heads) = 2048 floats / sample

typedef __attribute__((ext_vector_type(2))) float v2f;
typedef __attribute__((ext_vector_type(4))) float v4f;
typedef __attribute__((ext_vector_type(8))) float v8f;

// ---------------------------------------------------------------------------
// GEMM: logits[B, NOUT] = x[B, HIDDEN] @ W[HIDDEN, NOUT] + bias
// fp32 WMMA 16x16x4, 128x128 block tile, BK=16, 8 waves / block,
// double-buffered LDS. Wave grid 4(M) x 2(N); each wave owns 32x64.
//
// LDS layouts:
//   A: row-major [128][20]  (stride 20 dwords -> frag b64 reads tile all banks)
//   B: K-pair interleaved [8][288]: (k,n) -> [k>>1][2n + (k&1)]
//      so a WMMA B-fragment (K,K+1) is one aligned ds_load_b64;
//      stride 288 ≡ 32 (mod 64 banks) -> lane halves use disjoint banks.
// ---------------------------------------------------------------------------
#define BM 128
#define BN 128
#define BK 16
#define A_STRIDE (BK + 4)       // 20 dwords
#define BPS      (2 * BN + 32)  // 288 dwords per interleaved k-pair row
#define NITER    (HIDDEN / BK)  // 256

__global__ __launch_bounds__(256)
void gemm_logits_kernel(const float* __restrict__ X,
                        const float* __restrict__ W,
                        const float* __restrict__ bias,
                        float* __restrict__ logits) {
    __shared__ float As[2][BM * A_STRIDE];
    __shared__ float Bs[2][(BK / 2) * BPS];

    const int tid  = threadIdx.x;
    const int wid  = tid >> 5;
    const int lane = tid & 31;
    const int half = lane >> 4;   // 0: lanes 0-15, 1: lanes 16-31
    const int l16  = lane & 15;

    const int m_blk = blockIdx.y * BM;
    const int n_blk = blockIdx.x * BN;

    const int wm0 = (wid >> 1) * 32;  // wave M offset in block tile
    const int wn0 = (wid & 1) * 64;   // wave N offset in block tile

    v8f acc[2][4];
#pragma unroll
    for (int i = 0; i < 2; ++i)
#pragma unroll
        for (int j = 0; j < 4; ++j)
#pragma unroll
            for (int r = 0; r < 8; ++r) acc[i][j][r] = 0.0f;

    // A staging: 256 threads x 8 floats (two v4f), row-major
    const int a_row = tid >> 1;            // 0..127
    const int a_col = (tid & 1) * 8;       // 0 or 8
    // B staging: each thread loads 4 cols of a k-pair (two rows), interleaves
    const int kpair = tid >> 5;            // 0..7
    const int ncol  = (lane) * 4;          // 0..124

    const float* __restrict__ Xg = X + (size_t)(m_blk + a_row) * HIDDEN + a_col;
    const float* __restrict__ Wg = W + (size_t)(2 * kpair) * NOUT + (n_blk + ncol);

    // ---- prologue: stage tile 0 into buffer 0 ----
    {
        v4f xa0 = *(const v4f*)(Xg);
        v4f xa1 = *(const v4f*)(Xg + 4);
        v4f r0  = *(const v4f*)(Wg);
        v4f r1  = *(const v4f*)(Wg + NOUT);
        *(v4f*)&As[0][a_row * A_STRIDE + a_col]     = xa0;
        *(v4f*)&As[0][a_row * A_STRIDE + a_col + 4] = xa1;
        v4f s0 = {r0.x, r1.x, r0.y, r1.y};
        v4f s1 = {r0.z, r1.z, r0.w, r1.w};
        *(v4f*)&Bs[0][kpair * BPS + ncol * 2]     = s0;
        *(v4f*)&Bs[0][kpair * BPS + ncol * 2 + 4] = s1;
    }
    __syncthreads();

    for (int it = 0; it < NITER; ++it) {
        const int buf = it & 1;
        const int k0n = (it + 1) * BK;

        // ---- issue next tile's global loads (overlap with compute) ----
        v4f xa0, xa1, r0, r1;
        if (k0n < HIDDEN) {
            xa0 = *(const v4f*)(Xg + k0n);
            xa1 = *(const v4f*)(Xg + k0n + 4);
            r0  = *(const v4f*)(Wg + (size_t)k0n * NOUT);
            r1  = *(const v4f*)(Wg + (size_t)k0n * NOUT + NOUT);
            if (k0n + BK < HIDDEN) {  // warm L2 for tile after next
                __builtin_prefetch(Xg + k0n + BK, 0, 0);
                __builtin_prefetch(Wg + (size_t)(k0n + BK) * NOUT, 0, 0);
            }
        }

        // ---- compute from current buffer ----
#pragma unroll
        for (int kk = 0; kk < BK; kk += 4) {
            // A frag: lanes 0-15 hold K=kk,kk+1; lanes 16-31 hold K=kk+2,kk+3
            v2f afr[2];
#pragma unroll
            for (int i = 0; i < 2; ++i)
                afr[i] = *(const v2f*)
                    &As[buf][(wm0 + i * 16 + l16) * A_STRIDE + kk + 2 * half];
            // B frag: interleaved pair (K, K+1) at one b64 address
            const int kp = (kk >> 1) + half;
            v2f bfr[4];
#pragma unroll
            for (int j = 0; j < 4; ++j)
                bfr[j] = *(const v2f*)
                    &Bs[buf][kp * BPS + (wn0 + j * 16 + l16) * 2];
#pragma unroll
            for (int i = 0; i < 2; ++i)
#pragma unroll
                for (int j = 0; j < 4; ++j) {
#if __has_builtin(__builtin_amdgcn_wmma_f32_16x16x4_f32)
                    acc[i][j] = __builtin_amdgcn_wmma_f32_16x16x4_f32(
                        /*neg_a=*/false, afr[i], /*neg_b=*/false, bfr[j],
                        /*c_mod=*/(short)0, acc[i][j],
                        /*reuse_a=*/false, /*reuse_b=*/false);
#else
                    // layout-equivalent scalar fallback (also host pass)
                    const int mb = wm0 + i * 16 + half * 8;
#pragma unroll
                    for (int r = 0; r < 8; ++r) {
                        acc[i][j][r] += As[buf][(mb + r) * A_STRIDE + kk + 2 * half] * bfr[j].x;
                        acc[i][j][r] += As[buf][(mb + r) * A_STRIDE + kk + 2 * half + 1] * bfr[j].y;
                    }
#endif
                }
        }

        // ---- commit staged tile to the other buffer ----
        if (k0n < HIDDEN) {
            const int nb = buf ^ 1;
            *(v4f*)&As[nb][a_row * A_STRIDE + a_col]     = xa0;
            *(v4f*)&As[nb][a_row * A_STRIDE + a_col + 4] = xa1;
            v4f s0 = {r0.x, r1.x, r0.y, r1.y};
            v4f s1 = {r0.z, r1.z, r0.w, r1.w};
            *(v4f*)&Bs[nb][kpair * BPS + ncol * 2]     = s0;
            *(v4f*)&Bs[nb][kpair * BPS + ncol * 2 + 4] = s1;
        }
        __syncthreads();
    }

    // Store: D layout — lane half selects M+8; VGPR r = M row within half.
#pragma unroll
    for (int i = 0; i < 2; ++i) {
        const int row = m_blk + wm0 + i * 16 + half * 8;
#pragma unroll
        for (int j = 0; j < 4; ++j) {
            const int col = n_blk + wn0 + j * 16 + l16;
            const float bz = bias[col];
            float* __restrict__ op = logits + (size_t)row * NOUT + col;
#pragma unroll
            for (int r = 0; r < 8; ++r)
                op[(size_t)r * NOUT] = acc[i][j][r] + bz;
        }
    }
}

// ---------------------------------------------------------------------------
// Epilogue: gumbel-softmax, group scores, stable top-k vs budget, assemble
// output (prefix ones + straight-through hard/soft). One block per sample.
// ---------------------------------------------------------------------------
__global__ __launch_bounds__(256)
void scheduler_epilogue_kernel(const float* __restrict__ logits,
                               const float* __restrict__ latency,
                               const float* __restrict__ u,
                               float* __restrict__ out) {
    __shared__ float s_soft[NOUT];      // softmax probs, layout == logits row
    __shared__ float s_score[NGROUPS];  // group scores

    const int b   = blockIdx.x;
    const int tid = threadIdx.x;
    const float* __restrict__ lrow = logits + (size_t)b * NOUT;
    const float* __restrict__ urow = u + (size_t)b * NOUT;

    const float eps = 1e-6f;
    for (int g = tid; g < NGROUPS; g += blockDim.x) {
        float pk[2];
#pragma unroll
        for (int h = 0; h < 2; ++h) {
            const int n = 4 * g + 2 * h;
            const float u0 = urow[n] * (1.0f - 2.0f * eps) + eps;
            const float u1 = urow[n + 1] * (1.0f - 2.0f * eps) + eps;
            const float g0 = -logf(-logf(u0));
            const float g1 = -logf(-logf(u1));
            const float z0 = (lrow[n] + g0) / TAU_F;
            const float z1 = (lrow[n + 1] + g1) / TAU_F;
            const float m  = fmaxf(z0, z1);
            const float e0 = expf(z0 - m);
            const float e1 = expf(z1 - m);
            const float inv = 1.0f / (e0 + e1);
            const float s0 = e0 * inv;
            const float s1 = e1 * inv;
            s_soft[n]     = s0;
            s_soft[n + 1] = s1;
            pk[h] = s0;
        }
        s_score[g] = 0.5f * (pk[0] + pk[1]);
    }
    __syncthreads();

    // budget: units = clip(rint(latency*512), 128, 512); budget = units - 128
    const float budget =
        fminf(fmaxf(rintf(latency[b] * 512.0f), 128.0f), 512.0f) - 128.0f;

    float* __restrict__ orow = out + (size_t)b * OUT_PER_B;

    // prefix layers 0..7: all ones
    for (int i = tid; i < NUM_PREFIX * 2 * NHEADS; i += blockDim.x)
        orow[i] = 1.0f;

    // stable rank (argsort-of-argsort with stable ties == count of
    // strictly-greater plus equal-with-smaller-index), keep if rank < budget
    for (int g = tid; g < NGROUPS; g += blockDim.x) {
        const float sg = s_score[g];
        int cnt = 0;
        for (int j = 0; j < NGROUPS; ++j) {
            const float sj = s_score[j];
            cnt += (sj > sg) || (sj == sg && j < g);
        }
        const float hk = ((float)cnt < budget) ? 1.0f : 0.0f;

        const int layer = g >> 4;            // 16 groups per sub-layer
        const int h0    = (g & 15) * 2;      // first head in group
        float* __restrict__ lp = orow + (size_t)(NUM_PREFIX + layer) * 64;
#pragma unroll
        for (int h = 0; h < 2; ++h) {
            const int n = 4 * g + 2 * h;
            const float s0 = s_soft[n];
            const float s1 = s_soft[n + 1];
            // straight-through, reference FP order: hard + soft - soft
            lp[h0 + h]          = (hk + s0) - s0;
            lp[NHEADS + h0 + h] = ((1.0f - hk) + s1) - s1;
        }
    }
}

// ---------------------------------------------------------------------------
extern "C" void kernel_launch(void* const* d_in, const int* in_sizes, int n_in,
                              void* d_out, int out_size, void* d_ws, size_t ws_size,
                              hipStream_t stream) {
    (void)in_sizes; (void)n_in; (void)out_size; (void)ws_size;
    const float* x       = (const float*)d_in[0];
    const float* latency = (const float*)d_in[1];
    const float* W       = (const float*)d_in[2];
    const float* bias    = (const float*)d_in[3];
    const float* u       = (const float*)d_in[4];
    float* out    = (float*)d_out;
    float* logits = (float*)d_ws;   // BATCH * NOUT floats = 96 MiB scratch

    dim3 gemm_grid(NOUT / BN, BATCH / BM);   // N fastest -> x shared via L2
    gemm_logits_kernel<<<gemm_grid, 256, 0, stream>>>(x, W, bias, logits);

    scheduler_epilogue_kernel<<<BATCH, 256, 0, stream>>>(logits, latency, u, out);
}